// DKGA_Det_86105504350807
// MI455X (gfx1250) — compile-verified
//
#include <hip/hip_runtime.h>
#include <stdint.h>
#include <math.h>

// ---------------- constants ----------------
#define NB        16                 // batch
#define NTOT      87040u             // anchors per image (256^2+128^2+64^2+32^2)
#define NQUAD     (NTOT / 4u)        // 21760 anchor-quads per image
#define PRE       512                // pre-NMS top-k
#define CAP       1024               // gather capacity per image
#define NBUCK     32768              // histogram buckets (15-bit monotone key)
#define MAXD      300
#define SCORE_T   0.25f
#define NMS_T     0.45f

// workspace layout (bytes)
#define HIST_BYTES ((size_t)NB * NBUCK * 4)        // 2 MiB
#define CNT_OFF    (HIST_BYTES)
#define CUT_OFF    (CNT_OFF + 64)
#define CAND_OFF   (CUT_OFF + 64)
#define CAND_BYTES ((size_t)NB * CAP * 8)          // 128 KiB
#define WS_BYTES   (CAND_OFF + CAND_BYTES)
#define ZQUADS     ((uint32_t)(WS_BYTES / 16))     // WS_BYTES is 16B-multiple

// ---------------- helpers ----------------
__device__ __forceinline__ void lvl_of(uint32_t a, int& li, uint32_t& al,
                                       int& shift, float& stride, uint32_t& hw) {
  if (a < 65536u)      { li = 0; al = a;          shift = 8; stride = 8.f;  hw = 65536u; }
  else if (a < 81920u) { li = 1; al = a - 65536u; shift = 7; stride = 16.f; hw = 16384u; }
  else if (a < 86016u) { li = 2; al = a - 81920u; shift = 6; stride = 32.f; hw = 4096u;  }
  else                 { li = 3; al = a - 86016u; shift = 5; stride = 64.f; hw = 1024u;  }
}

__device__ __forceinline__ float sigmoid_stable(float x) {
  if (x >= 0.f) return 1.f / (1.f + expf(-x));
  float e = expf(x);
  return e / (1.f + e);
}

// monotone 15-bit key for scores in (0.25, 1): bits in (0x3E800000, 0x3F800000)
__device__ __forceinline__ uint32_t score_key(float s) {
  uint32_t b = __float_as_uint(s);
  if (b >= 0x3F800000u) return NBUCK - 1u;
  return (b - 0x3E800000u) >> 9;   // < 32768
}

__device__ __forceinline__ const float* pick4(int li, const float* p0, const float* p1,
                                              const float* p2, const float* p3) {
  return (li == 0) ? p0 : (li == 1) ? p1 : (li == 2) ? p2 : p3;
}

// ---------------- kernel 0: zero workspace (16B stores) ----------------
__global__ void det_zero_ws(uint4* w, uint32_t n) {
  uint32_t i = blockIdx.x * blockDim.x + threadIdx.x;
  if (i < n) w[i] = make_uint4(0u, 0u, 0u, 0u);
}

// ---------------- kernel 1: score histogram (4 anchors/thread) ----------------
__global__ void det_hist(const float* __restrict__ c0, const float* __restrict__ c1,
                         const float* __restrict__ c2, const float* __restrict__ c3,
                         uint32_t* __restrict__ hist) {
  uint32_t t = blockIdx.x * blockDim.x + threadIdx.x;
  if (t >= NB * NQUAD) return;
  uint32_t img = t / NQUAD, a0 = (t - img * NQUAD) * 4u;   // quad never straddles levels
  int li, shift; uint32_t al, hw; float stride;
  lvl_of(a0, li, al, shift, stride, hw);
  const float* cp = pick4(li, c0, c1, c2, c3) + (size_t)img * hw + al;
  __builtin_prefetch(cp + 4096, 0, 0);                     // global_prefetch_b8 (gfx1250)
  float4 v = *(const float4*)cp;
  uint32_t* h = hist + img * NBUCK;
  float s;
  s = sigmoid_stable(v.x); if (s > SCORE_T) atomicAdd(&h[score_key(s)], 1u);
  s = sigmoid_stable(v.y); if (s > SCORE_T) atomicAdd(&h[score_key(s)], 1u);
  s = sigmoid_stable(v.z); if (s > SCORE_T) atomicAdd(&h[score_key(s)], 1u);
  s = sigmoid_stable(v.w); if (s > SCORE_T) atomicAdd(&h[score_key(s)], 1u);
}

// ---------------- kernel 2: per-image cutoff key ----------------
__global__ __launch_bounds__(512) void det_cutoff(const uint32_t* __restrict__ hist,
                                                  uint32_t* __restrict__ cutoff) {
  __shared__ uint32_t sSum[512];
  __shared__ uint32_t sCut;
  int img = blockIdx.x, t = threadIdx.x;
  const uint32_t* h = hist + (size_t)img * NBUCK;
  // chunk t covers keys [hi-64, hi), chunks ordered from the TOP key downward
  int hi = NBUCK - 64 * t, lo = hi - 64;
  uint32_t s = 0;
  for (int k = lo; k < hi; k += 4) {
    uint4 q = *(const uint4*)&h[k];
    s += q.x + q.y + q.z + q.w;
  }
  sSum[t] = s;
  if (t == 0) sCut = 0u;
  __syncthreads();
  for (int off = 1; off < 512; off <<= 1) {      // inclusive Hillis-Steele scan
    uint32_t add = (t >= off) ? sSum[t - off] : 0u;
    __syncthreads();
    sSum[t] += add;
    __syncthreads();
  }
  uint32_t incl = sSum[t], excl = incl - s;
  if (excl < PRE && incl >= PRE) {               // crossing chunk: walk down
    uint32_t c = excl;
    for (int k = hi - 1; k >= lo; --k) {
      c += h[k];
      if (c >= PRE) { sCut = (uint32_t)k; break; }
    }
  }
  __syncthreads();
  if (t == 0) cutoff[img] = sCut;
}

// ---------------- kernel 3: gather candidates >= cutoff (4 anchors/thread) ----------------
__global__ void det_gather(const float* __restrict__ c0, const float* __restrict__ c1,
                           const float* __restrict__ c2, const float* __restrict__ c3,
                           const uint32_t* __restrict__ cutoff, uint32_t* __restrict__ cnt,
                           unsigned long long* __restrict__ cand) {
  uint32_t t = blockIdx.x * blockDim.x + threadIdx.x;
  if (t >= NB * NQUAD) return;
  uint32_t img = t / NQUAD, a0 = (t - img * NQUAD) * 4u;
  int li, shift; uint32_t al, hw; float stride;
  lvl_of(a0, li, al, shift, stride, hw);
  const float* cp = pick4(li, c0, c1, c2, c3) + (size_t)img * hw + al;
  __builtin_prefetch(cp + 4096, 0, 0);
  float4 v = *(const float4*)cp;
  uint32_t cut = cutoff[img];
  unsigned long long* cd = cand + (size_t)img * CAP;
  float sv[4] = {v.x, v.y, v.z, v.w};
  #pragma unroll
  for (int q = 0; q < 4; ++q) {
    float s = sigmoid_stable(sv[q]);
    if (s > SCORE_T && score_key(s) >= cut) {
      uint32_t pos = atomicAdd(&cnt[img], 1u);
      if (pos < CAP) {
        // key: score bits high (desc sort), ~index low (ties -> lowest index first)
        unsigned long long kv =
            ((unsigned long long)__float_as_uint(s) << 32) | (uint32_t)(~(a0 + q));
        cd[pos] = kv;
      }
    }
  }
}

// ---------------- kernel 4: sort + NMS + output ----------------
__global__ __launch_bounds__(512) void det_final(
    const float* __restrict__ r0, const float* __restrict__ r1,
    const float* __restrict__ r2, const float* __restrict__ r3,
    const float* __restrict__ k0, const float* __restrict__ k1,
    const float* __restrict__ k2, const float* __restrict__ k3,
    const unsigned long long* __restrict__ cand, float* __restrict__ out) {
  __shared__ unsigned long long sKey[CAP];   // 8 KiB
  __shared__ float4   sBox[PRE];             // 8 KiB
  __shared__ uint32_t sMask[PRE * 16];       // 32 KiB suppression-bit matrix
  __shared__ uint32_t sKeep[16];
  __shared__ uint32_t sPfx[16];

  const int img = blockIdx.x;
  const int t   = threadIdx.x;               // 512 threads
  float* outImg = out + (size_t)img * MAXD * 15;

  // zero this image's output region (buf starts as zeros in reference)
  for (int i = t; i < MAXD * 15; i += 512) outImg[i] = 0.f;

  // --- async DMA: stage this image's 1024 candidate keys into LDS (CDNA5 path)
  {
    unsigned ldsAddr = (unsigned)(uintptr_t)(&sKey[2 * t]);          // low 32 bits = LDS offset
    unsigned vOff    = (unsigned)t * 16u;                            // 16 B per thread
    unsigned long long base = (unsigned long long)(uintptr_t)(cand + (size_t)img * CAP);
    asm volatile("global_load_async_to_lds_b128 %0, %1, %2"
                 :: "v"(ldsAddr), "v"(vOff), "s"(base) : "memory");
    asm volatile("s_wait_asynccnt 0" ::: "memory");
  }
  __syncthreads();

  // --- bitonic sort 1024 u64 keys, descending
  for (unsigned k = 2; k <= CAP; k <<= 1) {
    for (unsigned j = k >> 1; j >= 1; j >>= 1) {
      __syncthreads();
      for (unsigned e = t; e < CAP; e += 512) {
        unsigned x = e ^ j;
        if (x > e) {
          unsigned long long va = sKey[e], vb = sKey[x];
          if ((va < vb) == ((e & k) == 0u)) { sKey[e] = vb; sKey[x] = va; }
        }
      }
    }
  }
  __syncthreads();

  // --- decode box for my sorted entry
  unsigned long long kv = sKey[t];
  float    score = __uint_as_float((uint32_t)(kv >> 32));
  uint32_t a     = ~(uint32_t)(kv & 0xFFFFFFFFull);
  bool valid = (score > SCORE_T) && (a < NTOT);
  float x1 = 0.f, y1 = 0.f, x2 = 0.f, y2 = 0.f, cx = 0.f, cy = 0.f, stride = 0.f;
  int li = 0, shift = 0; uint32_t al = 0, hw = 0;
  if (valid) {
    lvl_of(a, li, al, shift, stride, hw);
    const float* rp = pick4(li, r0, r1, r2, r3) + (size_t)(img * 4) * hw + al;
    float q0 = rp[0], q1 = rp[hw], q2 = rp[2 * (size_t)hw], q3 = rp[3 * (size_t)hw];
    uint32_t xq = al & ((1u << shift) - 1u), yq = al >> shift;
    cx = ((float)xq + 0.5f) * stride;
    cy = ((float)yq + 0.5f) * stride;
    float cxd = cx + q0 * stride, cyd = cy + q1 * stride;
    float w = expf(q2) * stride, h = expf(q3) * stride;
    x1 = cxd - 0.5f * w; y1 = cyd - 0.5f * h;
    x2 = cxd + 0.5f * w; y2 = cyd + 0.5f * h;
  }
  sBox[t] = make_float4(x1, y1, x2, y2);

  if (t < 16) sKeep[t] = 0u;
  __syncthreads();
  if (valid) atomicOr(&sKeep[t >> 5], 1u << (t & 31));

  // --- suppression-bit matrix: row t, bit j set iff j>t and IoU(t,j) > NMS_T
  __syncthreads();
  {
    float areaA = (x2 - x1) * (y2 - y1);
    for (int w = 0; w < 16; ++w) {
      uint32_t m = 0u;
      if (w * 32 + 31 > t) {                 // word contains some j > t
        int jb = w * 32;
        for (int jj = 0; jj < 32; ++jj) {
          int j = jb + jj;
          if (j > t) {
            float4 bb = sBox[j];
            float xx1 = fmaxf(x1, bb.x), yy1 = fmaxf(y1, bb.y);
            float xx2 = fminf(x2, bb.z), yy2 = fminf(y2, bb.w);
            float inter = fmaxf(xx2 - xx1, 0.f) * fmaxf(yy2 - yy1, 0.f);
            float areaB = (bb.z - bb.x) * (bb.w - bb.y);
            float uni = areaA + areaB - inter;
            float iou = inter / fmaxf(uni, 1e-9f);
            if (iou > NMS_T) m |= (1u << jj);
          }
        }
      }
      sMask[t * 16 + w] = m;
    }
  }
  __syncthreads();

  // --- greedy sweep: single wave32, 512-bit keep mask across 16 lanes, no barriers
  if (t < 32) {
    uint32_t kw = (t < 16) ? sKeep[t] : 0u;
    for (int i = 0; i < PRE; ++i) {
      uint32_t w = __shfl(kw, i >> 5, 32);          // broadcast word containing bit i
      if ((w >> (i & 31)) & 1u) {                   // i still kept -> suppress
        if (t < 16) kw &= ~sMask[i * 16 + t];
      }
    }
    if (t < 16) sKeep[t] = kw;
  }
  __syncthreads();

  if (t == 0) {
    uint32_t c = 0;
    for (int w = 0; w < 16; ++w) { sPfx[w] = c; c += __popc(sKeep[w]); }
  }
  __syncthreads();

  // --- compact kept detections into first MAXD rows (score order)
  bool kept = valid && ((sKeep[t >> 5] >> (t & 31)) & 1u);
  if (kept) {
    uint32_t rank = sPfx[t >> 5] + __popc(sKeep[t >> 5] & ((1u << (t & 31)) - 1u));
    if (rank < MAXD) {
      float* row = outImg + (size_t)rank * 15;
      row[0] = x1; row[1] = y1; row[2] = x2; row[3] = y2; row[4] = score;
      const float* kp = pick4(li, k0, k1, k2, k3) + (size_t)(img * 10) * hw + al;
      #pragma unroll
      for (int i = 0; i < 5; ++i) {
        row[5 + 2 * i] = cx + kp[(size_t)(2 * i) * hw] * stride;
        row[6 + 2 * i] = cy + kp[(size_t)(2 * i + 1) * hw] * stride;
      }
    }
  }
}

// ---------------- launch ----------------
extern "C" void kernel_launch(void* const* d_in, const int* in_sizes, int n_in,
                              void* d_out, int out_size, void* d_ws, size_t ws_size,
                              hipStream_t stream) {
  (void)in_sizes; (void)n_in; (void)out_size; (void)ws_size;
  // setup_inputs order: cls0,reg0,kpt0, cls1,reg1,kpt1, cls2,reg2,kpt2, cls3,reg3,kpt3
  const float* cls[4]; const float* reg[4]; const float* kpt[4];
  for (int l = 0; l < 4; ++l) {
    cls[l] = (const float*)d_in[3 * l + 0];
    reg[l] = (const float*)d_in[3 * l + 1];
    kpt[l] = (const float*)d_in[3 * l + 2];
  }
  uint8_t* ws = (uint8_t*)d_ws;
  uint32_t* hist = (uint32_t*)ws;
  uint32_t* cnt  = (uint32_t*)(ws + CNT_OFF);
  uint32_t* cut  = (uint32_t*)(ws + CUT_OFF);
  unsigned long long* cand = (unsigned long long*)(ws + CAND_OFF);
  float* out = (float*)d_out;

  det_zero_ws<<<(ZQUADS + 255) / 256, 256, 0, stream>>>((uint4*)ws, ZQUADS);

  uint32_t tot = NB * NQUAD;                       // 348,160 threads, 4 anchors each
  det_hist<<<(tot + 255) / 256, 256, 0, stream>>>(cls[0], cls[1], cls[2], cls[3], hist);
  det_cutoff<<<NB, 512, 0, stream>>>(hist, cut);
  det_gather<<<(tot + 255) / 256, 256, 0, stream>>>(cls[0], cls[1], cls[2], cls[3],
                                                    cut, cnt, cand);
  det_final<<<NB, 512, 0, stream>>>(reg[0], reg[1], reg[2], reg[3],
                                    kpt[0], kpt[1], kpt[2], kpt[3], cand, out);
}